// TALOSLoss_1151051236044
// MI455X (gfx1250) — compile-verified
//
#include <hip/hip_runtime.h>

typedef __attribute__((ext_vector_type(2))) float v2f;
typedef __attribute__((ext_vector_type(8))) float v8f;

#define MAX_SHIFT 15
#define NUM_SHIFTS 31
#define B_DIM 128
#define T_DIM 16384
#define CHUNK 1024
#define HALO 15

#if __has_builtin(__builtin_amdgcn_global_load_async_to_lds_b32) && \
    __has_builtin(__builtin_amdgcn_s_wait_asynccnt)
#define USE_ASYNC_LDS 1
typedef __attribute__((address_space(1))) int as1_int;
typedef __attribute__((address_space(3))) int as3_int;
#else
#define USE_ASYNC_LDS 0
#endif

// ---------------------------------------------------------------------------
// Workspace layout (floats):
//   [0, B*31)              corr[b][s]   (s = d + 15, d in [-15,15])
//   [B*31, B*31+B)         sx2[b] = sum_t y_pred^2
//   [B*31+B, B*31+2B)      sy2[b] = sum_t y_true^2
// ---------------------------------------------------------------------------

__global__ void talos_init(float* __restrict__ ws) {
  int i = blockIdx.x * blockDim.x + threadIdx.x;
  if (i < B_DIM * (NUM_SHIFTS + 2)) ws[i] = 0.0f;
}

// One wave32 per (batch, 1024-wide chunk). Computes all 31 lags of the
// cross-correlation via chained V_WMMA_F32_16X16X4_F32 using the Toeplitz
// anti-diagonal mapping:  D[m][n] accumulates lag d = m + n - 15.
__global__ __launch_bounds__(32) void talos_corr(const float* __restrict__ y_pred,
                                                 const float* __restrict__ y_true,
                                                 float* __restrict__ ws) {
  __shared__ float lds_x[CHUNK + 16]; // lds_x[i] = xpad[c0 + i - 15]
  __shared__ float lds_y[CHUNK + 16]; // lds_y[i] = ypad[c0 + i - 15]

  const int lane = threadIdx.x;      // 0..31 (wave32)
  const int b    = blockIdx.y;
  const int c0   = blockIdx.x * CHUNK;

  const float* __restrict__ xg = y_pred + (size_t)b * T_DIM;
  const float* __restrict__ yg = y_true + (size_t)b * T_DIM;

  float sx2 = 0.0f, sy2 = 0.0f;

#if USE_ASYNC_LDS
  // ---- stage chunk via GLOBAL_LOAD_ASYNC_TO_LDS_B32 (ASYNCcnt path) ----
  // Halo zeros (t < 0) only exist for chunk 0; DS-store them there so the
  // ds_store and async LDS writes never touch the same slots (unordered).
  if (c0 == 0 && lane < HALO) {
    lds_x[lane] = 0.0f;
    lds_y[lane] = 0.0f;
  }
  for (int i = lane; i < CHUNK + HALO; i += 32) {
    int g = c0 + i - HALO;                 // global t; g <= c0 + CHUNK - 1 < T
    if (g >= 0) {
      __builtin_amdgcn_global_load_async_to_lds_b32(
          (as1_int*)(xg + g), (as3_int*)&lds_x[i], 0, 0);
      __builtin_amdgcn_global_load_async_to_lds_b32(
          (as1_int*)(yg + g), (as3_int*)&lds_y[i], 0, 0);
    }
  }
  __builtin_amdgcn_s_wait_asynccnt(0);
  __syncthreads();
  // Squared-sum partials over chunk-owned t (re-read from LDS).
  for (int i = lane + HALO; i < CHUNK + HALO; i += 32) {
    float xv = lds_x[i];
    float yv = lds_y[i];
    sx2 += xv * xv;
    sy2 += yv * yv;
  }
#else
  // ---- fallback: load to VGPR, store to LDS ----
  for (int i = lane; i < CHUNK + HALO; i += 32) {
    int g = c0 + i - HALO;
    float xv = (g >= 0) ? xg[g] : 0.0f;
    float yv = (g >= 0) ? yg[g] : 0.0f;
    lds_x[i] = xv;
    lds_y[i] = yv;
    if (i >= HALO) { sx2 += xv * xv; sy2 += yv * yv; }
  }
  __syncthreads();
#endif

  // ---- WMMA lag accumulation ----
  // f32 16x16x4 operand layout (ISA 7.12.2):
  //   A (16x4):  lane L holds A[M][K0], A[M][K0+1]  with M = L&15, K0 = 2*(L>>4)
  //   B (4x16):  lane L holds B[K0][N], B[K0+1][N]  with N = L&15
  //   A[m][k] = xpad[s0 + k - m]      -> lds_x index = s0 + 15 + K0 - M
  //   B[k][n] = ypad[s0 + k + n - 15] -> lds_y index = s0 + K0 + N
  const int MN = lane & 15;
  const int K0 = (lane >> 4) << 1;
  const float* __restrict__ ax = &lds_x[HALO + K0 - MN];
  const float* __restrict__ by = &lds_y[K0 + MN];

  // Two independent accumulator chains so back-to-back WMMAs have no D->C RAW.
  v8f acc0 = {}, acc1 = {};
#pragma unroll 2
  for (int s0 = 0; s0 < CHUNK; s0 += 8) {
    v2f a0, b0, a1, b1;
    a0.x = ax[s0];     a0.y = ax[s0 + 1];
    b0.x = by[s0];     b0.y = by[s0 + 1];
    a1.x = ax[s0 + 4]; a1.y = ax[s0 + 5];
    b1.x = by[s0 + 4]; b1.y = by[s0 + 5];
    acc0 = __builtin_amdgcn_wmma_f32_16x16x4_f32(false, a0, false, b0,
                                                 (short)0, acc0, false, false);
    acc1 = __builtin_amdgcn_wmma_f32_16x16x4_f32(false, a1, false, b1,
                                                 (short)0, acc1, false, false);
  }
  v8f acc = acc0 + acc1;

  // ---- extract the 31 unique lags and atomically accumulate ----
  // C/D layout: VGPR v, lane<16 -> M=v, N=lane; lane>=16 -> M=v+8, N=lane-16.
  // d <= 0   : cell (m=0, n=d+15) -> lanes 0..15, acc[0]  -> corr[n]
  // d = 1..7 : cell (m=d, n=15)   -> lane 15, acc[1..7]   -> corr[16..22]
  // d = 8..15: cell (m=d, n=15)   -> lane 31, acc[0..7]   -> corr[23..30]
  float* __restrict__ corr = ws + (size_t)b * NUM_SHIFTS;
  if (lane < 16) atomicAdd(&corr[lane], acc[0]);
  if (lane == 15) {
    atomicAdd(&corr[16], acc[1]); atomicAdd(&corr[17], acc[2]);
    atomicAdd(&corr[18], acc[3]); atomicAdd(&corr[19], acc[4]);
    atomicAdd(&corr[20], acc[5]); atomicAdd(&corr[21], acc[6]);
    atomicAdd(&corr[22], acc[7]);
  }
  if (lane == 31) {
    atomicAdd(&corr[23], acc[0]); atomicAdd(&corr[24], acc[1]);
    atomicAdd(&corr[25], acc[2]); atomicAdd(&corr[26], acc[3]);
    atomicAdd(&corr[27], acc[4]); atomicAdd(&corr[28], acc[5]);
    atomicAdd(&corr[29], acc[6]); atomicAdd(&corr[30], acc[7]);
  }

  // ---- wave-reduce the squared sums ----
  for (int off = 16; off > 0; off >>= 1) {
    sx2 += __shfl_down(sx2, off, 32);
    sy2 += __shfl_down(sy2, off, 32);
  }
  if (lane == 0) {
    atomicAdd(&ws[B_DIM * NUM_SHIFTS + b], sx2);
    atomicAdd(&ws[B_DIM * NUM_SHIFTS + B_DIM + b], sy2);
  }
}

// One block, one thread per batch row: head/tail prefixes, softmax weights,
// MSE assembly, block-reduced mean.
__global__ __launch_bounds__(128) void talos_final(const float* __restrict__ y_true,
                                                   const float* __restrict__ theta,
                                                   const int* __restrict__ pidx,
                                                   const float* __restrict__ ws,
                                                   float* __restrict__ out) {
  __shared__ float red[B_DIM];
  const int b = threadIdx.x;

  const float* __restrict__ corr = ws + (size_t)b * NUM_SHIFTS;
  const float sx2 = ws[B_DIM * NUM_SHIFTS + b];
  const float sy2 = ws[B_DIM * NUM_SHIFTS + B_DIM + b];
  const float* __restrict__ yg = y_true + (size_t)b * T_DIM;

  // head[k] = sum_{t<k} y^2 ; tail[k] = sum_{t>=T-k} y^2  (k = 0..15)
  float head[MAX_SHIFT + 1], tail[MAX_SHIFT + 1];
  head[0] = 0.0f; tail[0] = 0.0f;
  for (int k = 1; k <= MAX_SHIFT; ++k) {
    float hv = yg[k - 1];
    float tv = yg[T_DIM - k];
    head[k] = head[k - 1] + hv * hv;
    tail[k] = tail[k - 1] + tv * tv;
  }

  const float* __restrict__ th = theta + (size_t)pidx[b] * NUM_SHIFTS;
  float mx = th[0];
  for (int s = 1; s < NUM_SHIFTS; ++s) mx = fmaxf(mx, th[s]);

  float den = 0.0f, num = 0.0f;
  for (int s = 0; s < NUM_SHIFTS; ++s) {
    int d = s - MAX_SHIFT;
    float sy2w = sy2 - ((d >= 0) ? head[d] : tail[-d]);
    float mse  = (sx2 + sy2w - 2.0f * corr[s]) * (1.0f / (float)T_DIM);
    float e    = expf(th[s] - mx);
    den += e;
    num += e * mse;
  }
  red[b] = num / den;
  __syncthreads();

  for (int off = 64; off > 0; off >>= 1) {
    if (b < off) red[b] += red[b + off];
    __syncthreads();
  }
  if (b == 0) out[0] = red[0] * (1.0f / (float)B_DIM);
}

extern "C" void kernel_launch(void* const* d_in, const int* in_sizes, int n_in,
                              void* d_out, int out_size, void* d_ws, size_t ws_size,
                              hipStream_t stream) {
  const float* y_pred = (const float*)d_in[0];
  const float* y_true = (const float*)d_in[1];
  const float* theta  = (const float*)d_in[2];
  const int*   pidx   = (const int*)d_in[3];
  float* ws  = (float*)d_ws;
  float* out = (float*)d_out;

  const int nws = B_DIM * (NUM_SHIFTS + 2);
  talos_init<<<(nws + 255) / 256, 256, 0, stream>>>(ws);

  dim3 grid(T_DIM / CHUNK, B_DIM);
  talos_corr<<<grid, 32, 0, stream>>>(y_pred, y_true, ws);

  talos_final<<<1, B_DIM, 0, stream>>>(y_true, theta, pidx, ws, out);
}